// AuxiliaryClustering_15796889715181
// MI455X (gfx1250) — compile-verified
//
#include <hip/hip_runtime.h>
#include <math.h>

typedef __attribute__((ext_vector_type(2))) float v2f;
typedef __attribute__((ext_vector_type(8))) float v8f;

#define KC 64          // number of clusters
#define DC 64          // latent dim
#define CSTR 68        // padded LDS stride for centers (68*4B: 16B aligned, bank-spreading)

// ---------------------------------------------------------------------------
// ws layout (floats):
//   [0   ..  63] column sums of assignments        (global atomic accum)
//   [64  .. 127] per-cluster distance sums         (global atomic accum)
//   [128 .. 191] per-cluster counts                (global atomic accum)
//   [192 .. 4287] 64x64 Gram matrix G = C * C^T    (written by WMMA kernel)
// ---------------------------------------------------------------------------

__global__ void init_ws_kernel(float* __restrict__ ws) {
    int i = blockIdx.x * blockDim.x + threadIdx.x;
    if (i < 192) ws[i] = 0.0f;
}

// Single streaming pass over both 256MB arrays: fused column-sum + argmax +
// distance-to-own-center. Memory-bound: reads each byte exactly once.
__global__ __launch_bounds__(256)
void main_pass_kernel(const float* __restrict__ z,
                      const float* __restrict__ a,
                      const float* __restrict__ c,
                      float* __restrict__ ws,
                      int B) {
    __shared__ float s_c[KC * CSTR];   // centers, padded stride
    __shared__ float s_col[KC];
    __shared__ float s_dsum[KC];
    __shared__ float s_cnt[KC];

    const int tid = threadIdx.x;

    // Stage centers into LDS (16KB payload), padded rows.
    for (int i = tid; i < KC * DC; i += 256) {
        int r = i >> 6, col = i & 63;
        s_c[r * CSTR + col] = c[i];
    }
    if (tid < KC) { s_col[tid] = 0.0f; s_dsum[tid] = 0.0f; s_cnt[tid] = 0.0f; }
    __syncthreads();

    // Register-resident column accumulators (constant-indexed -> stays in VGPRs).
    float colacc[KC];
#pragma unroll
    for (int k = 0; k < KC; ++k) colacc[k] = 0.0f;

    const long long stride = (long long)gridDim.x * (long long)blockDim.x;
    for (long long b = (long long)blockIdx.x * 256 + tid; b < B; b += stride) {
        // ---- assignments row: column sums + argmax (first-max semantics) ----
        const float4* arow = (const float4*)(a + b * (long long)KC);
        float best = -INFINITY;
        int   bi   = 0;
#pragma unroll
        for (int q = 0; q < 16; ++q) {
            float4 v = arow[q];
            colacc[4*q + 0] += v.x;
            colacc[4*q + 1] += v.y;
            colacc[4*q + 2] += v.z;
            colacc[4*q + 3] += v.w;
            if (v.x > best) { best = v.x; bi = 4*q + 0; }
            if (v.y > best) { best = v.y; bi = 4*q + 1; }
            if (v.z > best) { best = v.z; bi = 4*q + 2; }
            if (v.w > best) { best = v.w; bi = 4*q + 3; }
        }

        // ---- latent row: squared distance to own center ----
        const float4* zrow  = (const float4*)(z + b * (long long)DC);
        const float4* crow4 = (const float4*)(s_c + bi * CSTR);  // 272B stride, 16B aligned
        float sq = 0.0f;
#pragma unroll
        for (int q = 0; q < 16; ++q) {
            float4 v = zrow[q];
            float4 cc = crow4[q];
            float d0 = v.x - cc.x;
            float d1 = v.y - cc.y;
            float d2 = v.z - cc.z;
            float d3 = v.w - cc.w;
            sq += d0*d0 + d1*d1 + d2*d2 + d3*d3;
        }
        float dist = (sq > 0.0f) ? sqrtf(sq) : 0.0f;   // guarded sqrt (matches ref)
        atomicAdd(&s_dsum[bi], dist);                  // ds_add_f32
        atomicAdd(&s_cnt[bi], 1.0f);
    }

    // Flush per-thread column accumulators into LDS (constant indices).
#pragma unroll
    for (int k = 0; k < KC; ++k) atomicAdd(&s_col[k], colacc[k]);
    __syncthreads();

    if (tid < KC) {
        atomicAdd(&ws[tid],        s_col[tid]);
        atomicAdd(&ws[64  + tid],  s_dsum[tid]);
        atomicAdd(&ws[128 + tid],  s_cnt[tid]);
    }
}

// G = C * C^T (64x64, f32) via V_WMMA_F32_16X16X4_F32; single wave32.
// A fragment (16x4 MxK): lane(m = l&15, h = l>>4): {C[m][k+2h], C[m][k+2h+1]}
// B fragment (4x16 KxN): lane(n = l&15, h = l>>4): {B[k+2h][n], B[k+2h+1][n]}
//   and since B = C^T, B[k][n] = C[n][k]  ->  same gather as A with the tj tile.
// D (16x16 f32): VGPR j -> row (j + 8h), col (l&15).
__global__ __launch_bounds__(32)
void gram_wmma_kernel(const float* __restrict__ c, float* __restrict__ G) {
    const int lane = threadIdx.x;
    const int m = lane & 15;
    const int h = lane >> 4;

    for (int ti = 0; ti < 4; ++ti) {
        for (int tj = 0; tj < 4; ++tj) {
            v8f acc = {};
#pragma unroll
            for (int kk = 0; kk < 64; kk += 4) {
                v2f A, Bv;
                A.x  = c[(ti*16 + m) * 64 + kk + 2*h];
                A.y  = c[(ti*16 + m) * 64 + kk + 2*h + 1];
                Bv.x = c[(tj*16 + m) * 64 + kk + 2*h];
                Bv.y = c[(tj*16 + m) * 64 + kk + 2*h + 1];
                acc = __builtin_amdgcn_wmma_f32_16x16x4_f32(
                    /*neg_a=*/false, A, /*neg_b=*/false, Bv,
                    /*c_mod=*/(short)0, acc, /*reuse_a=*/false, /*reuse_b=*/false);
            }
#pragma unroll
            for (int j = 0; j < 8; ++j) {
                G[(ti*16 + j + 8*h) * 64 + tj*16 + m] = acc[j];
            }
        }
    }
}

__device__ __forceinline__ float block_reduce_256(float v, float* red, int tid) {
    red[tid] = v;
    __syncthreads();
    for (int s = 128; s > 0; s >>= 1) {
        if (tid < s) red[tid] += red[tid + s];
        __syncthreads();
    }
    float r = red[0];
    __syncthreads();
    return r;
}

__global__ __launch_bounds__(256)
void finalize_kernel(const float* __restrict__ ws, float* __restrict__ out, float Bf) {
    __shared__ float red[256];
    __shared__ float probs[KC];
    const int tid = threadIdx.x;

    if (tid < KC) probs[tid] = ws[tid] / Bf;
    __syncthreads();

    // --- balance: KL(uniform || probs), sum reduction ---
    float v = 0.0f;
    if (tid < KC) {
        const float u = 1.0f / 64.0f;
        v = u * (logf(u) - logf(probs[tid] + 1e-8f));
    }
    float balance = block_reduce_256(v, red, tid);

    // --- separation from Gram matrix ---
    const float* G = ws + 192;
    float ssum = 0.0f;
    for (int p = tid; p < 4096; p += 256) {
        int i = p >> 6, j = p & 63;
        if (i != j) {
            float sq = G[i*64 + i] + G[j*64 + j] - 2.0f * G[i*64 + j];
            ssum += (sq > 0.0f) ? sqrtf(sq) : 0.0f;
        }
    }
    float sepsum = block_reduce_256(ssum, red, tid);
    float separation = -sepsum / (64.0f * 63.0f);

    // --- compactness over non-empty clusters ---
    float cm = 0.0f, ne = 0.0f;
    if (tid < KC) {
        float cnt = ws[128 + tid];
        if (cnt > 0.0f) { cm = ws[64 + tid] / cnt; ne = 1.0f; }
    }
    float cmsum = block_reduce_256(cm, red, tid);
    float nesum = block_reduce_256(ne, red, tid);
    float compact = cmsum / fmaxf(nesum, 1.0f);

    // --- unbiased std of probs ---
    float psum = block_reduce_256(tid < KC ? probs[tid] : 0.0f, red, tid);
    float mean = psum / 64.0f;
    float dv = 0.0f;
    if (tid < KC) { float d = probs[tid] - mean; dv = d * d; }
    float sqsum = block_reduce_256(dv, red, tid);
    float stddev = sqrtf(sqsum / 63.0f);

    if (tid == 0) {
        float aux = 0.1f * balance + 0.1f * separation + 0.1f * compact;
        out[0] = aux;
        out[1] = balance;
        out[2] = separation;
        out[3] = compact;
        out[4] = stddev;
    }
}

extern "C" void kernel_launch(void* const* d_in, const int* in_sizes, int n_in,
                              void* d_out, int out_size, void* d_ws, size_t ws_size,
                              hipStream_t stream) {
    const float* z = (const float*)d_in[0];   // [B, 64]
    const float* a = (const float*)d_in[1];   // [B, 64]
    const float* c = (const float*)d_in[2];   // [64, 64]
    float* out = (float*)d_out;               // 5 scalars
    float* ws  = (float*)d_ws;

    const int B = in_sizes[0] / DC;

    init_ws_kernel<<<1, 256, 0, stream>>>(ws);

    const int rows_per_thread = 8;
    int blocks = (B + 256 * rows_per_thread - 1) / (256 * rows_per_thread);
    if (blocks < 1) blocks = 1;
    main_pass_kernel<<<blocks, 256, 0, stream>>>(z, a, c, ws, B);

    gram_wmma_kernel<<<1, 32, 0, stream>>>(c, ws + 192);

    finalize_kernel<<<1, 256, 0, stream>>>(ws, out, (float)B);
}